// FPSkNN_13159779795314
// MI455X (gfx1250) — compile-verified
//
#include <hip/hip_runtime.h>

#define NPTS   8192
#define NBATCH 16
#define NS     512
#define KNN    32
#define NFEAT  128

typedef float v2f __attribute__((ext_vector_type(2)));
typedef float v8f __attribute__((ext_vector_type(8)));

// ---------------------------------------------------------------------------
// Kernel 1: Farthest Point Sampling. One block per batch; per-thread point
// coords + running min-distance live entirely in registers (32 pts/thread).
// Argmax with first-index tie-break via monotone u64 key (d >= 0 always).
// ---------------------------------------------------------------------------
#define FPS_T 256
#define PPT   (NPTS / FPS_T)   // 32

__global__ __launch_bounds__(FPS_T) void fps_kernel(const float* __restrict__ xyz,
                                                    int* __restrict__ fps_idx) {
  __shared__ unsigned long long redbuf[FPS_T / 32];
  __shared__ int s_far;
  const int b   = blockIdx.x;
  const int tid = threadIdx.x;
  const float* base = xyz + (size_t)b * NPTS * 3;

  float px[PPT], py[PPT], pz[PPT], dist[PPT];
#pragma unroll
  for (int i = 0; i < PPT; ++i) {
    int p = tid + i * FPS_T;
    px[i] = base[p * 3 + 0];
    py[i] = base[p * 3 + 1];
    pz[i] = base[p * 3 + 2];
    dist[i] = 1e10f;
  }

  int far = 0;
  for (int s = 0; s < NS; ++s) {
    if (tid == 0) fps_idx[b * NS + s] = far;   // reference emits carry BEFORE update
    float cx = base[far * 3 + 0];
    float cy = base[far * 3 + 1];
    float cz = base[far * 3 + 2];

    float bestd = -1.0f;
    int   besti = 0;
#pragma unroll
    for (int i = 0; i < PPT; ++i) {
      float dx = px[i] - cx, dy = py[i] - cy, dz = pz[i] - cz;
      float d = dx * dx + dy * dy + dz * dz;
      d = fminf(dist[i], d);
      dist[i] = d;
      if (d > bestd) { bestd = d; besti = tid + i * FPS_T; }  // ascending idx => first max kept
    }
    // key: larger dist wins; on tie, smaller index wins (N-1-idx maximized)
    unsigned long long key = ((unsigned long long)__float_as_uint(bestd) << 32) |
                             (unsigned)(NPTS - 1 - besti);
#pragma unroll
    for (int off = 16; off > 0; off >>= 1) {
      unsigned long long o = __shfl_xor(key, off);
      key = (o > key) ? o : key;
    }
    if ((tid & 31) == 0) redbuf[tid >> 5] = key;
    __syncthreads();
    if (tid == 0) {
      unsigned long long k = redbuf[0];
#pragma unroll
      for (int w = 1; w < FPS_T / 32; ++w) k = (redbuf[w] > k) ? redbuf[w] : k;
      s_far = NPTS - 1 - (int)(unsigned)(k & 0xffffffffu);
    }
    __syncthreads();
    far = s_far;
  }
}

// ---------------------------------------------------------------------------
// Kernel 2: gather local coordinates + local features. One wave per (b,s).
// ---------------------------------------------------------------------------
__global__ __launch_bounds__(256) void gather_local(const float* __restrict__ xyz,
                                                    const float* __restrict__ feat,
                                                    const int* __restrict__ fps_idx,
                                                    float* __restrict__ out_lc,
                                                    float* __restrict__ out_lf) {
  const int g    = blockIdx.x * 8 + (threadIdx.x >> 5);   // 0 .. NBATCH*NS-1
  const int lane = threadIdx.x & 31;
  const int b    = g / NS;
  const int pi   = fps_idx[g];
  const float* src = xyz + ((size_t)b * NPTS + pi) * 3;
  if (lane < 3) out_lc[(size_t)g * 3 + lane] = src[lane];
  const float4* f  = (const float4*)(feat + ((size_t)b * NPTS + pi) * NFEAT);
  float4*       of = (float4*)(out_lf + (size_t)g * NFEAT);
  of[lane] = f[lane];                                      // 32 lanes x float4 = 128 floats
}

// ---------------------------------------------------------------------------
// Kernel 3: kNN. 16x16 distance tiles via V_WMMA_F32_16X16X4_F32:
//   A[m,:] = (-2qx, -2qy, -2qz, |q|^2)   B[:,n] = (x, y, z, 1)
//   C[m,n] = |q|^2 - 2 q.x ;  d = C + |x|^2   (same expansion as reference)
// A layout (32b 16x4): lanes 0-15 hold K=0(V0),K=1(V1); lanes 16-31 K=2,K=3.
// B layout (32b 4x16): lanes 0-15 rows K=0(V0),K=1(V1); lanes 16-31 K=2,K=3.
// C layout: VGPR r, lane L -> M = r + 8*(L>=16), N = L&15.
// Fragment builds use scalar ternaries (v_cndmask) so the float4 LDS read
// stays a single unconditional ds_load_b128 -- no EXEC-masked load splitting.
// ---------------------------------------------------------------------------
#define QPB 128   // queries per block (== threads per block)
#define KT  128
#define CH  64    // points per chunk
#define CHP 65    // LDS row stride (conflict-free consume)

__global__ __launch_bounds__(KT) void knn_kernel(const float* __restrict__ xyz,
                                                 const int* __restrict__ fps_idx,
                                                 int* __restrict__ knn_idx,
                                                 float* __restrict__ out_kc) {
  extern __shared__ float smem[];
  float4* pts   = (float4*)smem;        // NPTS entries: (x,y,z,|x|^2)  = 128 KB
  float*  dtile = smem + NPTS * 4;      // QPB * CHP floats             = 32.5 KB

  const int b   = blockIdx.x / (NS / QPB);
  const int qg  = blockIdx.x % (NS / QPB);
  const int tid = threadIdx.x;
  const float* base = xyz + (size_t)b * NPTS * 3;

  for (int p = tid; p < NPTS; p += KT) {
    float x = base[p * 3 + 0], y = base[p * 3 + 1], z = base[p * 3 + 2];
    float4 v; v.x = x; v.y = y; v.z = z; v.w = x * x + y * y + z * z;
    pts[p] = v;
  }
  __syncthreads();

  const int  lane = tid & 31;
  const int  wave = tid >> 5;        // 0..3
  const bool hi   = lane >= 16;
  const int  lm   = lane & 15;

  // Loop-invariant A fragments: this wave owns query rows wave*32 .. wave*32+31
  v2f afrag[2];
#pragma unroll
  for (int t = 0; t < 2; ++t) {
    int q  = qg * QPB + wave * 32 + t * 16 + lm;
    int pi = fps_idx[b * NS + q];
    float4 c = pts[pi];
    v2f a;
    a.x = hi ? (-2.0f * c.z) : (-2.0f * c.x);   // K=2 (z)      | K=0 (x)
    a.y = hi ? c.w           : (-2.0f * c.y);   // K=3 (|q|^2)  | K=1 (y)
    afrag[t] = a;
  }

  // per-thread sorted top-32 (ascending dist; tie -> lower index), register-resident
  float bd[KNN]; int bi[KNN];
#pragma unroll
  for (int j = 0; j < KNN; ++j) { bd[j] = 1e30f; bi[j] = 0x7fffffff; }

  for (int p0 = 0; p0 < NPTS; p0 += CH) {
    // --- compute phase: each wave fills its 32 query rows x 64 point cols ---
#pragma unroll
    for (int pt = 0; pt < CH / 16; ++pt) {
      int pb = p0 + pt * 16;
      float4 cp = pts[pb + lm];            // single ds_load_b128, lanes L/L+16 broadcast
      v2f bfrag;
      bfrag.x = hi ? cp.z : cp.x;          // row K=2 (z)    | K=0 (x)
      bfrag.y = hi ? 1.0f : cp.y;          // row K=3 (ones) | K=1 (y)
      float px2 = cp.w;                    // |x_n|^2 for column n = lm
#pragma unroll
      for (int qt = 0; qt < 2; ++qt) {
        v8f acc = {};
        acc = __builtin_amdgcn_wmma_f32_16x16x4_f32(
            /*neg_a=*/false, afrag[qt], /*neg_b=*/false, bfrag,
            /*c_mod=*/(short)0, acc, /*reuse_a=*/false, /*reuse_b=*/false);
        int qbase = wave * 32 + qt * 16 + (hi ? 8 : 0);
#pragma unroll
        for (int r = 0; r < 8; ++r)
          dtile[(qbase + r) * CHP + pt * 16 + lm] = acc[r] + px2;
      }
    }
    __syncthreads();
    // --- consume phase: thread tid owns query row tid ---
    const float* row = dtile + tid * CHP;
    for (int j = 0; j < CH; ++j) {
      float d = row[j];
      if (d < bd[KNN - 1]) {
        bd[KNN - 1] = d; bi[KNN - 1] = p0 + j;
#pragma unroll
        for (int m = KNN - 1; m > 0; --m) {
          bool sw = (bd[m] < bd[m - 1]) ||
                    (bd[m] == bd[m - 1] && bi[m] < bi[m - 1]);
          if (sw) {
            float td = bd[m]; bd[m] = bd[m - 1]; bd[m - 1] = td;
            int   ti = bi[m]; bi[m] = bi[m - 1]; bi[m - 1] = ti;
          }
        }
      }
    }
    __syncthreads();
  }

  // emit knn indices (workspace) + knn coordinates (from LDS cache)
  const int q = qg * QPB + tid;
  int*   oi = knn_idx + ((size_t)b * NS + q) * KNN;
  float* oc = out_kc + ((size_t)b * NS + q) * KNN * 3;
#pragma unroll
  for (int j = 0; j < KNN; ++j) {
    int pi = bi[j];
    oi[j] = pi;
    float4 c = pts[pi];
    oc[j * 3 + 0] = c.x; oc[j * 3 + 1] = c.y; oc[j * 3 + 2] = c.z;
  }
}

// ---------------------------------------------------------------------------
// Kernel 4: gather knn features. One wave per (b,s,k) row; float4 coalesced.
// ---------------------------------------------------------------------------
__global__ __launch_bounds__(256) void gather_knn_feat(const float* __restrict__ feat,
                                                       const int* __restrict__ knn_idx,
                                                       float* __restrict__ out_kf) {
  const int g    = blockIdx.x * 8 + (threadIdx.x >> 5);  // 0 .. NBATCH*NS*KNN-1
  const int lane = threadIdx.x & 31;
  const int b    = g / (NS * KNN);
  const int pi   = knn_idx[g];
  const float4* f  = (const float4*)(feat + ((size_t)b * NPTS + pi) * NFEAT);
  float4*       of = (float4*)(out_kf + (size_t)g * NFEAT);
  of[lane] = f[lane];
}

// ---------------------------------------------------------------------------
extern "C" void kernel_launch(void* const* d_in, const int* in_sizes, int n_in,
                              void* d_out, int out_size, void* d_ws, size_t ws_size,
                              hipStream_t stream) {
  const float* xyz  = (const float*)d_in[0];   // (16, 8192, 3)
  const float* feat = (const float*)d_in[1];   // (16, 8192, 128)

  float* out    = (float*)d_out;
  float* out_lc = out;                                         // (16,512,3)
  float* out_lf = out_lc + (size_t)NBATCH * NS * 3;            // (16,512,128)
  float* out_kc = out_lf + (size_t)NBATCH * NS * NFEAT;        // (16,512,32,3)
  float* out_kf = out_kc + (size_t)NBATCH * NS * KNN * 3;      // (16,512,32,128)

  int* fps_idx = (int*)d_ws;                 // 16*512 ints   (32 KB)
  int* knn_idx = fps_idx + NBATCH * NS;      // 16*512*32 ints (1 MB)

  fps_kernel<<<NBATCH, FPS_T, 0, stream>>>(xyz, fps_idx);

  gather_local<<<(NBATCH * NS) / 8, 256, 0, stream>>>(xyz, feat, fps_idx,
                                                      out_lc, out_lf);

  size_t lds = (size_t)NPTS * sizeof(float4) + (size_t)QPB * CHP * sizeof(float);
  knn_kernel<<<NBATCH * (NS / QPB), KT, lds, stream>>>(xyz, fps_idx, knn_idx, out_kc);

  gather_knn_feat<<<(NBATCH * NS * KNN) / 8, 256, 0, stream>>>(feat, knn_idx, out_kf);
}